// ChunkDropout_13864154431504
// MI455X (gfx1250) — compile-verified
//
#include <hip/hip_runtime.h>
#include <stdint.h>

// ---------------------------------------------------------------------------
// ChunkDropout on MI455X (gfx1250)
//   out[b,l] = x[b,l] * keep[l] * scale      (f32, 4096 x 10000)
// Bandwidth-bound: ~327.7 MB traffic -> ~14us floor @ 23.3 TB/s.
// Kernel A: sequential jax-threefry mask generation (1 block).
// Kernel B: streaming multiply; keep tile staged to LDS via TDM
//           (tensor_load_to_lds + s_wait_tensorcnt), x/out via NT b128.
// ---------------------------------------------------------------------------

#define L_LEN        10000
#define L_PAD        10240   // padded so TDM tile reads stay in-bounds
#define BATCH        4096
#define COLS4        2500    // 10000 / 4
#define ROWS_PER_BLK 16
#define TILE_F4      256     // float4 columns per block (256 threads)
#define TILE_F       1024    // floats per keep tile

typedef __attribute__((ext_vector_type(4))) float        v4f;
typedef __attribute__((ext_vector_type(4))) unsigned int v4u;
typedef __attribute__((ext_vector_type(8))) int          v8i;
typedef __attribute__((ext_vector_type(4))) int          v4i;

// ---------------- threefry2x32 (jax PRNG core) -----------------------------
__device__ __forceinline__ uint32_t rotl32(uint32_t x, int r) {
    return (x << r) | (x >> (32 - r));
}

__device__ __forceinline__ void threefry2x32(uint32_t k0, uint32_t k1,
                                             uint32_t c0, uint32_t c1,
                                             uint32_t* o0, uint32_t* o1) {
    const uint32_t ks0 = k0, ks1 = k1, ks2 = k0 ^ k1 ^ 0x1BD11BDAu;
    const uint32_t ks[3] = {ks0, ks1, ks2};
    const int rotA[4] = {13, 15, 26, 6};
    const int rotB[4] = {17, 29, 16, 24};
    uint32_t x0 = c0 + ks0, x1 = c1 + ks1;
    #pragma unroll
    for (int i = 0; i < 5; ++i) {
        const int* r = (i & 1) ? rotB : rotA;
        #pragma unroll
        for (int j = 0; j < 4; ++j) {
            x0 += x1; x1 = rotl32(x1, r[j]); x1 ^= x0;
        }
        x0 += ks[(i + 1) % 3];
        x1 += ks[(i + 2) % 3] + (uint32_t)(i + 1);
    }
    *o0 = x0; *o1 = x1;
}

// XLA f32 erf_inv (Giles polynomial) — matches jax.random.normal
__device__ __forceinline__ float erfinv_f32(float x) {
    float w = -logf((1.0f - x) * (1.0f + x));
    float p;
    if (w < 5.0f) {
        w -= 2.5f;
        p =               2.81022636e-08f;
        p = fmaf(p, w,    3.43273939e-07f);
        p = fmaf(p, w,   -3.5233877e-06f);
        p = fmaf(p, w,   -4.39150654e-06f);
        p = fmaf(p, w,    0.00021858087f);
        p = fmaf(p, w,   -0.00125372503f);
        p = fmaf(p, w,   -0.00417768164f);
        p = fmaf(p, w,    0.246640727f);
        p = fmaf(p, w,    1.50140941f);
    } else {
        w = sqrtf(w) - 3.0f;
        p =              -0.000200214257f;
        p = fmaf(p, w,    0.000100950558f);
        p = fmaf(p, w,    0.00134934322f);
        p = fmaf(p, w,   -0.00367342844f);
        p = fmaf(p, w,    0.00573950773f);
        p = fmaf(p, w,   -0.0076224613f);
        p = fmaf(p, w,    0.00943887047f);
        p = fmaf(p, w,    1.00167406f);
        p = fmaf(p, w,    2.83297682f);
    }
    return p * x;
}

// bits -> float in [0,1): bitcast((b>>9)|0x3f800000) - 1
__device__ __forceinline__ float bits_to_unit(uint32_t b) {
    return __uint_as_float((b >> 9) | 0x3f800000u) - 1.0f;
}

// ---------------- Kernel A: sequential mask generation ---------------------
__global__ __launch_bounds__(256)
void chunkdropout_mask_kernel(const float* __restrict__ scaling,
                              float* __restrict__ keepScaled) {
    __shared__ unsigned char hole[L_LEN];
    for (int i = threadIdx.x; i < L_LEN; i += 256) hole[i] = 0;
    __syncthreads();

    if (threadIdx.x == 0) {
        // jax.random.key(42) -> (0, 42)
        uint32_t k0 = 0u, k1 = 42u;
        const float log1mp = logf(1.0f - 0.01f);
        const float tiny   = 1.17549435e-38f;
        const float lo_n   = -0.99999994f;   // nextafter(-1, 0) f32
        int  last_end = 0;
        bool done = false;
        while (!done) {
            // key, kg, kn = split(key, 3): counts iota(6) split [0,1,2],[3,4,5]
            uint32_t a0, b0, a1, b1, a2, b2;
            threefry2x32(k0, k1, 0u, 3u, &a0, &b0);
            threefry2x32(k0, k1, 1u, 4u, &a1, &b1);
            threefry2x32(k0, k1, 2u, 5u, &a2, &b2);
            k0 = a0; k1 = a1;                  // new key
            const uint32_t g0k = a2, g1k = b0; // kg
            const uint32_t n0k = b1, n1k = b2; // kn

            // uniform(kg, (), f32, tiny, 1.0): bits = threefry(kg, (0,0)).x
            uint32_t r0, r1;
            threefry2x32(g0k, g1k, 0u, 0u, &r0, &r1);
            float u = bits_to_unit(r0) * (1.0f - tiny) + tiny;
            u = fmaxf(u, tiny);
            int offset = (int)floorf(logf(u) / log1mp);
            if (offset < 1) offset = 1;
            const int  gap_start   = last_end + offset;
            const bool stop_before = (gap_start >= L_LEN - 1);

            // normal(kn): u in [nextafter(-1,0), 1); sqrt(2)*erfinv(u)
            threefry2x32(n0k, n1k, 0u, 0u, &r0, &r1);
            float un = bits_to_unit(r0) * (1.0f - lo_n) + lo_n;
            un = fmaxf(un, lo_n);
            const float nrm = 1.4142135623730951f * erfinv_f32(un);
            int glen = (int)(nrm * 3.0f + 10.0f);   // trunc-toward-zero
            if (glen < 1) glen = 1;
            int gap_end = gap_start + glen;
            if (gap_end > L_LEN) gap_end = L_LEN;

            if (!stop_before) {
                for (int i = gap_start; i < gap_end; ++i) hole[i] = 1;
                last_end = gap_end;
            }
            done = stop_before || (gap_end >= L_LEN);
        }
    }
    __syncthreads();

    const float s = scaling[0];
    for (int i = threadIdx.x; i < L_PAD; i += 256) {
        float v = 0.0f;
        if (i < L_LEN) v = hole[i] ? 0.0f : s;
        keepScaled[i] = v;
    }
}

// ---------------- Kernel B: streaming multiply with TDM-staged keep --------
__global__ __launch_bounds__(256)
void chunkdropout_scale_kernel(const float* __restrict__ x,
                               const float* __restrict__ keepScaled,
                               float* __restrict__ out) {
    __shared__ float keepTile[TILE_F];

    const int tid      = threadIdx.x;
    const int tileCol4 = blockIdx.x * TILE_F4;         // float4 column base
    const int rowBase  = blockIdx.y * ROWS_PER_BLK;

    // Wave 0 issues one TDM DMA for the 4KB keep tile -> LDS, then waits.
    if (tid < 32) {
        const unsigned long long gaddr =
            (unsigned long long)(uintptr_t)(keepScaled + (size_t)tileCol4 * 4);
        const unsigned int ldsAddr =
            (unsigned int)(uintptr_t)(void*)keepTile;  // flat LDS addr[31:0]

        // D# group 0: count=1 | lds_addr | global_addr[56:0] | type=2
        v4u g0;
        g0.x = 1u;                                          // count=1, user
        g0.y = ldsAddr;                                     // lds_addr
        g0.z = (unsigned int)(gaddr & 0xFFFFFFFFu);         // gaddr[31:0]
        g0.w = (unsigned int)((gaddr >> 32) & 0x01FFFFFFu)  // gaddr[56:32]
             | (2u << 30);                                  // type=2 (image)

        // D# group 1: data_size=4B, tensor 10240x1, tile 1024x1, stride 10240
        v8i g1;
        g1[0] = (int)(2u << 16);            // [17:16] data_size = 2 (4 bytes)
        g1[1] = (int)(10240u << 16);        // tensor_dim0[15:0] @ [63:48]
        g1[2] = (int)(1u << 16);            // tensor_dim0 hi=0 | tensor_dim1=1
        g1[3] = (int)(1024u << 16);         // tile_dim0=1024 @ [127:112]
        g1[4] = 1;                          // tile_dim1=1, tile_dim2=0
        g1[5] = 10240;                      // tensor_dim0_stride[31:0]
        g1[6] = 0;                          // stride hi / dim1_stride
        g1[7] = 0;
        v4i g2 = {0, 0, 0, 0};
        v4i g3 = {0, 0, 0, 0};
        v8i g4 = {0, 0, 0, 0, 0, 0, 0, 0};

        __builtin_amdgcn_tensor_load_to_lds(g0, g1, g2, g3, g4, 0);
        __builtin_amdgcn_s_wait_tensorcnt(0);
    }
    __syncthreads();

    const int c4 = tileCol4 + tid;
    if (c4 < COLS4) {
        const v4f k = ((const v4f*)keepTile)[tid];   // ds_load_b128
        const v4f* __restrict__ xin  = (const v4f*)x;
        v4f*       __restrict__ xout = (v4f*)out;
        long base = (long)rowBase * COLS4 + c4;
        #pragma unroll
        for (int r = 0; r < ROWS_PER_BLK; ++r) {
            v4f xv = __builtin_nontemporal_load(&xin[base]);   // NT b128 load
            v4f ov = xv * k;
            __builtin_nontemporal_store(ov, &xout[base]);      // NT b128 store
            base += COLS4;
        }
    }
}

// ---------------------------------------------------------------------------
extern "C" void kernel_launch(void* const* d_in, const int* in_sizes, int n_in,
                              void* d_out, int out_size, void* d_ws, size_t ws_size,
                              hipStream_t stream) {
    const float* x       = (const float*)d_in[0];
    const float* scaling = (const float*)d_in[1];
    float*       out     = (float*)d_out;
    float*       keep    = (float*)d_ws;   // 10240 floats (40 KB) of scratch

    hipLaunchKernelGGL(chunkdropout_mask_kernel, dim3(1), dim3(256), 0, stream,
                       scaling, keep);

    dim3 grid((COLS4 + TILE_F4 - 1) / TILE_F4,   // 10 column tiles
              BATCH / ROWS_PER_BLK);             // 256 row groups
    hipLaunchKernelGGL(chunkdropout_scale_kernel, grid, dim3(256), 0, stream,
                       x, keep, out);
}